// Graph_56478819943000
// MI455X (gfx1250) — compile-verified
//
#include <hip/hip_runtime.h>
#include <hip/hip_bf16.h>

// ---------------------------------------------------------------------------
// CDNA5 / gfx1250: 3-layer K-hop GraphConv GNN.
// Dense math: V_WMMA_F32_16X16X4_F32 (exact fp32), LDS double-buffered block
// GEMM with async global->LDS staging (ASYNCcnt path) when available.
// ---------------------------------------------------------------------------

typedef __attribute__((ext_vector_type(2))) float v2f;
typedef __attribute__((ext_vector_type(8))) float v8f;
typedef __attribute__((ext_vector_type(4))) int   v4i;

#define N_NODES  20000
#define N_GRAPHS 64

#define MBLK 64
#define NBLK 64
#define KC   32
#define LDA  36   // padded LDS stride (floats) for A tile (144B: 16B mult, bank-stagger)
#define LDB  68   // padded LDS stride (floats) for B tile (272B)
#define ASZ  (MBLK * LDA)   // 2304 floats
#define BSZ  (KC * LDB)     // 2176 floats

#if __has_builtin(__builtin_amdgcn_global_load_async_to_lds_b128)
#define HAVE_ASYNC_LDS 1
#endif

// 16-byte global -> LDS stage; async engine (ASYNCcnt) if the toolchain
// exposes it.  Builtin signature (from clang diagnostics): param0 is the
// global source as v4i in address space 1, param1 the LDS dest (AS 3).
__device__ __forceinline__ void stage16(const float* g, float* l)
{
#ifdef HAVE_ASYNC_LDS
    __builtin_amdgcn_global_load_async_to_lds_b128(
        (__attribute__((address_space(1))) v4i*)g,
        (__attribute__((address_space(3))) v4i*)l,
        0, 0);
#else
    *(float4*)l = *(const float4*)g;
#endif
}

__device__ __forceinline__ void stage_fence()
{
#ifdef HAVE_ASYNC_LDS
    asm volatile("s_wait_asynccnt 0x0" ::: "memory");
#endif
}

// ---------------------------------------------------------------------------
// Fused K-hop GEMM:  C = bias + sum_{k=0..3} H_k @ W_k   (+ optional LeakyReLU)
// 256 threads = 8 waves; block computes a 64x64 C macro-tile.  Wave w owns
// A-row-block mi = w&3 and N-half nh = w>>2 (two 16x16 subtiles sharing one
// A fragment).  K consumed in KC=32 chunks, double-buffered through LDS:
// while chunk t is computed (8 fp32 WMMAs per subtile pair), chunk t+1 is
// staged into the other buffer; one s_wait_asynccnt + barrier per chunk.
// EXEC stays all-1s at every WMMA (ragged last M-block handled by clamping
// A-row loads; only the final stores are predicated).
// ---------------------------------------------------------------------------
__global__ __launch_bounds__(256) void khop_gemm_wmma(
    const float* __restrict__ h0, const float* __restrict__ h1,
    const float* __restrict__ h2, const float* __restrict__ h3,
    const float* __restrict__ W,     // [4, K, N] row-major
    const float* __restrict__ bias,  // [N]
    float* __restrict__ C,           // [M, N]
    int M, int K, int N, int leaky)
{
    __shared__ float sA[2 * ASZ];
    __shared__ float sB[2 * BSZ];

    const int nBlocks = N / NBLK;
    const int mb = blockIdx.x / nBlocks;
    const int nb = blockIdx.x - mb * nBlocks;
    const int mBase = mb * MBLK;
    const int nBase = nb * NBLK;

    const int tid  = threadIdx.x;
    const int wave = tid >> 5;
    const int lane = tid & 31;
    const int mi   = wave & 3;          // A row block (16 rows)
    const int nh   = wave >> 2;         // N half (32 cols)
    const int hi   = lane >> 4;         // lane half
    const int lo   = lane & 15;
    const int nc   = nh * 32 + lo;      // local col of first subtile

    const float* Hs[4] = { h0, h1, h2, h3 };
    const int kchunks = K / KC;         // chunks per hop matrix
    const int T = 4 * kchunks;          // total chunks

    // issue the staging loads of chunk t into buffer b (no waits here)
    auto issueStage = [&](int t, int b) {
        const int km = t / kchunks;
        const int kk = (t - km * kchunks) * KC;
        const float* __restrict__ H  = Hs[km];
        const float* __restrict__ Wk = W + (size_t)km * K * N;
        float* __restrict__ dA = sA + b * ASZ;
        float* __restrict__ dB = sB + b * BSZ;
        // A: 64 rows x 32 cols (512 float4 / 256 threads)
#pragma unroll
        for (int i = 0; i < 2; ++i) {
            int linear = tid + 256 * i;
            int row = linear >> 3;
            int col = (linear & 7) << 2;
            int gm  = mBase + row; if (gm > M - 1) gm = M - 1;   // clamp ragged block
            stage16(H + (size_t)gm * K + kk + col, dA + row * LDA + col);
        }
        // B: 32 rows x 64 cols
#pragma unroll
        for (int i = 0; i < 2; ++i) {
            int linear = tid + 256 * i;
            int row = linear >> 4;
            int col = (linear & 15) << 2;
            stage16(Wk + (size_t)(kk + row) * N + nBase + col, dB + row * LDB + col);
        }
    };

    v8f c0, c1;
    {
        const float bv0 = bias[nBase + nc];
        const float bv1 = bias[nBase + nc + 16];
#pragma unroll
        for (int r = 0; r < 8; ++r) { c0[r] = bv0; c1[r] = bv1; }
    }

    // prologue: stage chunk 0 into buffer 0
    issueStage(0, 0);
    stage_fence();
    __syncthreads();

    for (int t = 0; t < T; ++t) {
        const int cur = t & 1;
        if (t + 1 < T) issueStage(t + 1, cur ^ 1);   // overlap next stage

        const float* __restrict__ aRow = sA + cur * ASZ + (mi * 16 + lo) * LDA;
        const float* __restrict__ bBuf = sB + cur * BSZ;
#pragma unroll
        for (int k4 = 0; k4 < KC; k4 += 4) {
            const int ka = k4 + 2 * hi;
            v2f a = *(const v2f*)(aRow + ka);
            v2f b0, b1;
            b0[0] = bBuf[ka * LDB + nc];
            b0[1] = bBuf[(ka + 1) * LDB + nc];
            b1[0] = bBuf[ka * LDB + nc + 16];
            b1[1] = bBuf[(ka + 1) * LDB + nc + 16];
            c0 = __builtin_amdgcn_wmma_f32_16x16x4_f32(
                     false, a, false, b0, (short)0, c0, false, false);
            c1 = __builtin_amdgcn_wmma_f32_16x16x4_f32(
                     false, a, false, b1, (short)0, c1, false, false);
        }
        stage_fence();          // my async writes into the other buffer landed
        __syncthreads();        // everyone done reading cur / writing next
    }

    // ---- epilogue: LeakyReLU + guarded store -----------------------------
#pragma unroll
    for (int r = 0; r < 8; ++r) {
        int m = mBase + mi * 16 + r + 8 * hi;
        if (m < M) {
            float v0 = c0[r], v1 = c1[r];
            if (leaky) {
                v0 = v0 > 0.0f ? v0 : 0.01f * v0;
                v1 = v1 > 0.0f ? v1 : 0.01f * v1;
            }
            C[(size_t)m * N + nBase + nc]      = v0;
            C[(size_t)m * N + nBase + nc + 16] = v1;
        }
    }
}

// ---------------------------------------------------------------------------
// One message-passing hop, float4-vectorized:
//   dst_h[dst[e], f..f+3] += src_h[src[e], f..f+3]
// ---------------------------------------------------------------------------
__global__ void scatter_add_hop(const float* __restrict__ src_h,
                                float* __restrict__ dst_h,
                                const int* __restrict__ ei,  // [2, nEdges]
                                int nEdges, int F)
{
    const int F4 = F >> 2;
    long long total  = (long long)nEdges * F4;
    long long idx    = (long long)blockIdx.x * blockDim.x + threadIdx.x;
    long long stride = (long long)gridDim.x * blockDim.x;
    for (; idx < total; idx += stride) {
        int e = (int)(idx / F4);
        int f = (int)(idx - (long long)e * F4) << 2;
        int s = ei[e];
        int d = ei[nEdges + e];
        float4 v = *(const float4*)(src_h + (size_t)s * F + f);
        float* p = dst_h + (size_t)d * F + f;
        atomicAdd(p + 0, v.x);
        atomicAdd(p + 1, v.y);
        atomicAdd(p + 2, v.z);
        atomicAdd(p + 3, v.w);
    }
}

__global__ void fill_kernel(float* __restrict__ p, float v, long long n)
{
    long long idx    = (long long)blockIdx.x * blockDim.x + threadIdx.x;
    long long stride = (long long)gridDim.x * blockDim.x;
    for (; idx < n; idx += stride) p[idx] = v;
}

// float atomic max via int/uint min-max trick (valid for mixed signs)
__device__ __forceinline__ void atomicMaxF(float* addr, float val)
{
    if (val >= 0.0f) atomicMax((int*)addr, __float_as_int(val));
    else             atomicMin((unsigned int*)addr, __float_as_uint(val));
}

__global__ void pool_max_kernel(const float* __restrict__ H,
                                const int* __restrict__ batch,
                                float* __restrict__ P, int M, int F)
{
    const int F4 = F >> 2;
    long long total  = (long long)M * F4;
    long long idx    = (long long)blockIdx.x * blockDim.x + threadIdx.x;
    long long stride = (long long)gridDim.x * blockDim.x;
    for (; idx < total; idx += stride) {
        int m = (int)(idx / F4);
        int f = (int)(idx - (long long)m * F4) << 2;
        float4 v = *(const float4*)(H + (size_t)m * F + f);
        float* p = P + (size_t)batch[m] * F + f;
        atomicMaxF(p + 0, v.x);
        atomicMaxF(p + 1, v.y);
        atomicMaxF(p + 2, v.z);
        atomicMaxF(p + 3, v.w);
    }
}

// Tiny classifier layers: thread-per-output dot product (negligible cost).
__global__ void dense_kernel(const float* __restrict__ A,
                             const float* __restrict__ W,
                             const float* __restrict__ b,
                             float* __restrict__ C,
                             int M, int K, int N, int relu)
{
    int idx = blockIdx.x * blockDim.x + threadIdx.x;
    if (idx >= M * N) return;
    int m = idx / N, n = idx - m * N;
    float acc = b[n];
    const float* a = A + (size_t)m * K;
    for (int k = 0; k < K; ++k) acc = fmaf(a[k], W[(size_t)k * N + n], acc);
    if (relu) acc = fmaxf(acc, 0.0f);
    C[idx] = acc;
}

// ---------------------------------------------------------------------------
extern "C" void kernel_launch(void* const* d_in, const int* in_sizes, int n_in,
                              void* d_out, int out_size, void* d_ws, size_t ws_size,
                              hipStream_t stream)
{
    const float* x     = (const float*)d_in[0];
    const int*   ei    = (const int*)  d_in[1];
    const int*   batch = (const int*)  d_in[2];
    const float* W1 = (const float*)d_in[3];  const float* b1 = (const float*)d_in[4];
    const float* W2 = (const float*)d_in[5];  const float* b2 = (const float*)d_in[6];
    const float* W3 = (const float*)d_in[7];  const float* b3 = (const float*)d_in[8];
    const float* Wm1 = (const float*)d_in[9];  const float* bm1 = (const float*)d_in[10];
    const float* Wm2 = (const float*)d_in[11]; const float* bm2 = (const float*)d_in[12];
    const float* Wm3 = (const float*)d_in[13]; const float* bm3 = (const float*)d_in[14];

    const int NN = N_NODES;
    const int NE = in_sizes[1] / 2;
    const int NG = N_GRAPHS;

    // ---- workspace layout (floats) --------------------------------------
    float* ws    = (float*)d_ws;
    float* OUT_A = ws;                          // 20000*512 (layer1 out / layer3 out)
    float* OUT_B = OUT_A + (size_t)NN * 512;    // 20000*256 (layer2 out)
    float* H1    = OUT_B + (size_t)NN * 256;    // hop buffers, 20000*256 each
    float* H2    = H1    + (size_t)NN * 256;
    float* H3    = H2    + (size_t)NN * 256;
    float* POOL  = H3    + (size_t)NN * 256;    // 64*512
    float* T1    = POOL  + (size_t)NG * 512;    // 64*1024
    float* T2    = T1    + (size_t)NG * 1024;   // 64*512

    auto hops = [&](const float* h0, int F) {
        size_t bytes  = (size_t)NN * F * sizeof(float);
        int    blocks = (int)(((long long)NE * (F / 4) + 255) / 256);
        (void)hipMemsetAsync(H1, 0, bytes, stream);
        scatter_add_hop<<<blocks, 256, 0, stream>>>(h0, H1, ei, NE, F);
        (void)hipMemsetAsync(H2, 0, bytes, stream);
        scatter_add_hop<<<blocks, 256, 0, stream>>>(H1, H2, ei, NE, F);
        (void)hipMemsetAsync(H3, 0, bytes, stream);
        scatter_add_hop<<<blocks, 256, 0, stream>>>(H2, H3, ei, NE, F);
    };
    auto gemm = [&](const float* h0, const float* W, const float* b,
                    float* Cout, int K, int N, int leaky) {
        int grid = ((NN + MBLK - 1) / MBLK) * (N / NBLK);
        khop_gemm_wmma<<<grid, 256, 0, stream>>>(h0, H1, H2, H3, W, b, Cout,
                                                 NN, K, N, leaky);
    };

    // layer 1: 128 -> 128, LeakyReLU
    hops(x, 128);
    gemm(x, W1, b1, OUT_A, 128, 128, 1);
    // layer 2: 128 -> 256, LeakyReLU
    hops(OUT_A, 128);
    gemm(OUT_A, W2, b2, OUT_B, 128, 256, 1);
    // layer 3: 256 -> 512
    hops(OUT_B, 256);
    gemm(OUT_B, W3, b3, OUT_A, 256, 512, 0);

    // global max pool per graph
    fill_kernel<<<(NG * 512 + 255) / 256, 256, 0, stream>>>(POOL, -3.402823466e38f,
                                                            (long long)NG * 512);
    pool_max_kernel<<<(int)(((long long)NN * 128 + 255) / 256), 256, 0, stream>>>(
        OUT_A, batch, POOL, NN, 512);

    // classifier MLP
    dense_kernel<<<(NG * 1024 + 255) / 256, 256, 0, stream>>>(POOL, Wm1, bm1, T1, NG, 512, 1024, 1);
    dense_kernel<<<(NG * 512  + 255) / 256, 256, 0, stream>>>(T1,   Wm2, bm2, T2, NG, 1024, 512, 1);
    dense_kernel<<<1, 256, 0, stream>>>(T2, Wm3, bm3, (float*)d_out, NG, 512, 4, 0);
}